// SynapsePredictor_13073880449661
// MI455X (gfx1250) — compile-verified
//
#include <hip/hip_runtime.h>

#define N_NODES 100000
#define N_EDGES 1600000
#define N_PAIRS 200000
#define CH      128

typedef __attribute__((ext_vector_type(2))) float v2f;
typedef __attribute__((ext_vector_type(8))) float v8f;
typedef __attribute__((ext_vector_type(4))) int   v4i;

#define AS1 __attribute__((address_space(1)))
#define AS3 __attribute__((address_space(3)))

// ---------------------------------------------------------------------------
// Zero a float buffer (graph-capture safe; no hipMemset needed)
// ---------------------------------------------------------------------------
__global__ void k_zero(float* __restrict__ p, long n) {
  long i = (long)blockIdx.x * blockDim.x + threadIdx.x;
  long stride = (long)gridDim.x * blockDim.x;
  for (; i < n; i += stride) p[i] = 0.0f;
}

// ---------------------------------------------------------------------------
// In-degree count: cnt[dst[e]] += 1  (native f32 atomic add)
// ---------------------------------------------------------------------------
__global__ void k_degree(const int* __restrict__ dst, float* __restrict__ cnt) {
  int e = blockIdx.x * blockDim.x + threadIdx.x;
  if (e < N_EDGES) unsafeAtomicAdd(&cnt[dst[e]], 1.0f);
}

// ---------------------------------------------------------------------------
// Edge scatter: agg[dst] += x[src] * w   (32 lanes per edge, float4 chunks)
// Roofline-dominant phase: 819 MB gather + 819 MB atomic per layer.
// unsafeAtomicAdd forces single-instruction global_atomic_add_f32.
// ---------------------------------------------------------------------------
__global__ __launch_bounds__(256)
void k_scatter(const float* __restrict__ xin, const int* __restrict__ src,
               const int* __restrict__ dst, const float* __restrict__ ew,
               float* __restrict__ agg) {
  int tid = blockIdx.x * blockDim.x + threadIdx.x;
  int e  = tid >> 5;   // 32 lanes per edge
  int c4 = tid & 31;   // float4 chunk index (32 * 4 = 128 channels)
  if (e >= N_EDGES) return;
  int s = src[e], d = dst[e];
  float w = ew[e];
  const float4* x4 = (const float4*)xin;
  float4 v = x4[(long)s * 32 + c4];
  float* a = agg + (long)d * CH + c4 * 4;
  unsafeAtomicAdd(a + 0, v.x * w);
  unsafeAtomicAdd(a + 1, v.y * w);
  unsafeAtomicAdd(a + 2, v.z * w);
  unsafeAtomicAdd(a + 3, v.w * w);
}

// ---------------------------------------------------------------------------
// Combine: out = (agg / max(cnt,1)) @ w_rel + bias + xin @ w_root  [+ relu]
// f32 WMMA 16x16x4. One block = one 16-row strip; 8 waves = 8 col tiles.
// Tiles are contiguous 8KB chunks -> staged with async-to-LDS b128 DMA
// (ASYNCcnt path); mean-normalization is folded into the A fragment per
// lane (A-row M = lane&15).
//
// A-operand (16x4 f32): lanes 0-15 hold K={kb,kb+1}, lanes 16-31 K={kb+2,kb+3},
// M = lane&15.  B-operand mirrors: per-lane K pair from rows of W, N = lane&15.
// C/D: VGPR v -> M = v + 8*(lane>=16), N = lane&15.
// ---------------------------------------------------------------------------
template <bool RELU>
__global__ __launch_bounds__(256)
void k_combine(const float* __restrict__ agg, const float* __restrict__ cnt,
               const float* __restrict__ xin,
               const float* __restrict__ w_rel, const float* __restrict__ bias,
               const float* __restrict__ w_root, float* __restrict__ out) {
  __shared__ __align__(16) float sAgg[16 * CH];   // raw aggregate tile (8 KB)
  __shared__ __align__(16) float sX[16 * CH];     // root-input tile   (8 KB)

  const int tid = threadIdx.x;
  const int rowBase = blockIdx.x * 16;
  const long base = (long)rowBase * CH;

#if __has_builtin(__builtin_amdgcn_global_load_async_to_lds_b128)
  // 512 x 16B transfers per tile, 2 per thread per tile, no VGPR round-trip.
  for (int i = tid; i < 512; i += 256) {
    __builtin_amdgcn_global_load_async_to_lds_b128(
        (AS1 v4i*)(agg + base + (long)i * 4), (AS3 v4i*)(sAgg + i * 4), 0, 0);
    __builtin_amdgcn_global_load_async_to_lds_b128(
        (AS1 v4i*)(xin + base + (long)i * 4), (AS3 v4i*)(sX + i * 4), 0, 0);
  }
#if __has_builtin(__builtin_amdgcn_s_wait_asynccnt)
  __builtin_amdgcn_s_wait_asynccnt(0);
#else
  asm volatile("s_wait_asynccnt 0x0" ::: "memory");
#endif
#else
  for (int i = tid; i < 16 * CH; i += 256) {
    sAgg[i] = agg[base + i];
    sX[i]   = xin[base + i];
  }
#endif
  __syncthreads();

  const int wave = tid >> 5;
  const int lane = tid & 31;
  const int n    = lane & 15;   // also A-matrix row M for this lane
  const int half = lane >> 4;   // lane group selects K pair
  const int koff = half * 2;
  const int colBase = wave * 16;

  // mean-normalization factor for this lane's A row
  const float inv = 1.0f / fmaxf(cnt[rowBase + n], 1.0f);

  v8f acc = {};
  for (int kb = 0; kb < CH; kb += 4) {
    const int k0 = kb + koff;
    v2f aA, aX, bR, bT;
    aA.x = sAgg[n * CH + k0] * inv;
    aA.y = sAgg[n * CH + k0 + 1] * inv;
    aX.x = sX[n * CH + k0];
    aX.y = sX[n * CH + k0 + 1];
    bR.x = w_rel[(long)k0 * CH + colBase + n];
    bR.y = w_rel[(long)(k0 + 1) * CH + colBase + n];
    bT.x = w_root[(long)k0 * CH + colBase + n];
    bT.y = w_root[(long)(k0 + 1) * CH + colBase + n];
    acc = __builtin_amdgcn_wmma_f32_16x16x4_f32(false, aA, false, bR,
                                                (short)0, acc, false, false);
    acc = __builtin_amdgcn_wmma_f32_16x16x4_f32(false, aX, false, bT,
                                                (short)0, acc, false, false);
  }

  const float b = bias[colBase + n];
  for (int v = 0; v < 8; ++v) {
    int m = v + half * 8;
    float val = acc[v] + b;
    if (RELU) val = fmaxf(val, 0.0f);
    out[(long)(rowBase + m) * CH + colBase + n] = val;
  }
}

// ---------------------------------------------------------------------------
// Decode: out[p] = dot(z[s], z[d]) / sqrt(128).  One wave32 per pair.
// ---------------------------------------------------------------------------
__global__ __launch_bounds__(256)
void k_decode(const float* __restrict__ z, const int* __restrict__ eli,
              float* __restrict__ out) {
  int tid = blockIdx.x * blockDim.x + threadIdx.x;
  int pair = tid >> 5;
  int lane = tid & 31;
  if (pair >= N_PAIRS) return;
  int s = eli[pair];
  int d = eli[N_PAIRS + pair];
  const float4* z4 = (const float4*)z;
  float4 a = z4[(long)s * 32 + lane];
  float4 b = z4[(long)d * 32 + lane];
  float p = a.x * b.x + a.y * b.y + a.z * b.z + a.w * b.w;
  for (int off = 16; off; off >>= 1) p += __shfl_xor(p, off, 32);
  if (lane == 0) out[pair] = p * 0.08838834764831845f;  // 1/sqrt(128)
}

// ---------------------------------------------------------------------------
extern "C" void kernel_launch(void* const* d_in, const int* in_sizes, int n_in,
                              void* d_out, int out_size, void* d_ws, size_t ws_size,
                              hipStream_t stream) {
  const float* x       = (const float*)d_in[0];
  const int*   ei      = (const int*)d_in[1];
  const float* ew      = (const float*)d_in[2];
  const int*   eli     = (const int*)d_in[3];
  const float* w1_rel  = (const float*)d_in[4];
  const float* b1      = (const float*)d_in[5];
  const float* w1_root = (const float*)d_in[6];
  const float* w2_rel  = (const float*)d_in[7];
  const float* b2      = (const float*)d_in[8];
  const float* w2_root = (const float*)d_in[9];
  float* out = (float*)d_out;

  // Workspace layout (floats): agg[N*CH] | h[N*CH] | z[N*CH] | cnt[N]
  float* agg = (float*)d_ws;
  float* h   = agg + (size_t)N_NODES * CH;
  float* z   = h + (size_t)N_NODES * CH;
  float* cnt = z + (size_t)N_NODES * CH;

  const int* src = ei;             // edge_index[0]
  const int* dst = ei + N_EDGES;   // edge_index[1]

  const int scatterBlocks = (N_EDGES * 32) / 256;  // 200000
  const int combineBlocks = N_NODES / 16;          // 6250 (exact)

  // ---- layer 1 ----
  k_zero<<<2048, 256, 0, stream>>>(agg, (long)N_NODES * CH);
  k_zero<<<256, 256, 0, stream>>>(cnt, (long)N_NODES);
  k_degree<<<(N_EDGES + 255) / 256, 256, 0, stream>>>(dst, cnt);
  k_scatter<<<scatterBlocks, 256, 0, stream>>>(x, src, dst, ew, agg);
  k_combine<true><<<combineBlocks, 256, 0, stream>>>(agg, cnt, x, w1_rel, b1,
                                                     w1_root, h);
  // ---- layer 2 ----
  k_zero<<<2048, 256, 0, stream>>>(agg, (long)N_NODES * CH);
  k_scatter<<<scatterBlocks, 256, 0, stream>>>(h, src, dst, ew, agg);
  k_combine<false><<<combineBlocks, 256, 0, stream>>>(agg, cnt, h, w2_rel, b2,
                                                      w2_root, z);
  // ---- decode ----
  k_decode<<<(N_PAIRS * 32 + 255) / 256, 256, 0, stream>>>(z, eli, out);
}